// NGCFLayer_6614249636664
// MI455X (gfx1250) — compile-verified
//
#include <hip/hip_runtime.h>
#include <math.h>

#define DD 64
#define LSTR 66   // padded LDS row stride (floats) to break bank conflicts

typedef float v2f __attribute__((ext_vector_type(2)));
typedef float v8f __attribute__((ext_vector_type(8)));

// ---------------------------------------------------------------------------
// Phase 1: edge scatter.  64 threads per edge (one per feature dim).
// Accumulates (fits entirely in 192MB L2):
//   aggI1[dst] += n_ui * fu      aggI2[dst] += n_ui * (fu*fi)     sI[dst] += n_ui
//   aggU1[src] += n_iu * fi      aggU2[src] += n_iu * (fu*fi)     sU[src] += n_iu
// ---------------------------------------------------------------------------
__global__ __launch_bounds__(256) void ngcf_scatter(
    const float* __restrict__ feat_user, const float* __restrict__ feat_item,
    const float* __restrict__ norm_ui,   const float* __restrict__ norm_iu,
    const int*   __restrict__ ui_src,    const int*   __restrict__ ui_dst,
    float* __restrict__ aggU1, float* __restrict__ aggU2,
    float* __restrict__ aggI1, float* __restrict__ aggI2,
    float* __restrict__ sU,    float* __restrict__ sI, int E)
{
    long gid = (long)blockIdx.x * blockDim.x + threadIdx.x;
    int  e   = (int)(gid >> 6);
    int  d   = (int)(gid & 63);
    if (e >= E) return;

    int   src = ui_src[e];
    int   dst = ui_dst[e];
    float nu  = norm_ui[e];
    float ni  = norm_iu[e];

    // streaming reads: non-temporal so they don't evict L2-resident accumulators
    float fu = __builtin_nontemporal_load(&feat_user[(size_t)src * DD + d]);
    float fi = __builtin_nontemporal_load(&feat_item[(size_t)dst * DD + d]);
    float pr = fu * fi;

    atomicAdd(&aggI1[(size_t)dst * DD + d], nu * fu);
    atomicAdd(&aggI2[(size_t)dst * DD + d], nu * pr);
    atomicAdd(&aggU1[(size_t)src * DD + d], ni * fi);
    atomicAdd(&aggU2[(size_t)src * DD + d], ni * pr);
    if (d == 0) {
        atomicAdd(&sI[dst], nu);
        atomicAdd(&sU[src], ni);
    }
}

// ---------------------------------------------------------------------------
// Phase 2: fused dual-GEMM + bias + LeakyReLU + row L2-normalize, via
// V_WMMA_F32_16X16X4_F32 (full fp32 fidelity on the matrix pipe).
// Block = 256 threads = 8 wave32s, handles 32 node-rows:
//   wave w -> row group g=w>>2 (16 rows), col tile t=w&3 (16 cols of 64).
// A/B fragments come from LDS tiles (padded stride).
// ---------------------------------------------------------------------------
__global__ __launch_bounds__(256) void ngcf_gemm(
    const float* __restrict__ feat_user, const float* __restrict__ feat_item,
    const float* __restrict__ W1g, const float* __restrict__ b1g,
    const float* __restrict__ W2g, const float* __restrict__ b2g,
    const float* __restrict__ aggU1, const float* __restrict__ aggU2,
    const float* __restrict__ aggI1, const float* __restrict__ aggI2,
    const float* __restrict__ sU,    const float* __restrict__ sI,
    float* __restrict__ out, int NU, int Ntot)
{
    __shared__ float W1s[DD * LSTR];
    __shared__ float W2s[DD * LSTR];
    __shared__ float Xs [32 * LSTR];   // feat + agg1 tile
    __shared__ float Ps [32 * LSTR];   // agg2 tile
    __shared__ float b1s[DD], b2s[DD];
    __shared__ float Ss  [32];         // per-row norm sums
    __shared__ float rowss[32];        // per-row sum of squares (post-activation)

    const int tid  = threadIdx.x;
    const int row0 = blockIdx.x * 32;

    // cooperative load of weights / biases
    for (int i = tid; i < DD * DD; i += 256) {
        int r = i >> 6, c = i & 63;
        W1s[r * LSTR + c] = W1g[i];
        W2s[r * LSTR + c] = W2g[i];
    }
    if (tid < DD) { b1s[tid] = b1g[tid]; b2s[tid] = b2g[tid]; }

    // cooperative load of this block's 32 rows (user or item selected per row)
    for (int i = tid; i < 32 * DD; i += 256) {
        int r = i >> 6, c = i & 63;
        int row = row0 + r;
        float x = 0.f, p = 0.f;
        if (row < NU) {
            size_t o = (size_t)row * DD + c;
            x = feat_user[o] + aggU1[o];
            p = aggU2[o];
        } else if (row < Ntot) {
            size_t o = (size_t)(row - NU) * DD + c;
            x = feat_item[o] + aggI1[o];
            p = aggI2[o];
        }
        Xs[r * LSTR + c] = x;
        Ps[r * LSTR + c] = p;
    }
    if (tid < 32) {
        int row = row0 + tid;
        float s = 0.f;
        if (row < NU)        s = sU[row];
        else if (row < Ntot) s = sI[row - NU];
        Ss[tid]    = s;
        rowss[tid] = 0.f;
    }
    __syncthreads();

    const int wave = tid >> 5;
    const int lane = tid & 31;
    const int g    = wave >> 2;      // row group (0..1)
    const int t    = wave & 3;       // col tile (0..3)
    const int half = lane >> 4;      // K sub-select per WMMA layout
    const int l15  = lane & 15;

    const float* Xrow  = &Xs [(g * 16 + l15) * LSTR];   // A: M = l15
    const float* Prow  = &Ps [(g * 16 + l15) * LSTR];
    const float* W1row = &W1s[(t * 16 + l15) * LSTR];   // B: N = l15 (B[k][n] = W[n][k])
    const float* W2row = &W2s[(t * 16 + l15) * LSTR];

    v8f acc = {};
#pragma unroll
    for (int kk = 0; kk < DD; kk += 4) {
        const int ko = kk + 2 * half;   // lanes 0-15 -> K,K+1 ; lanes 16-31 -> K+2,K+3
        v2f a1 = *(const v2f*)(Xrow  + ko);
        v2f w1 = *(const v2f*)(W1row + ko);
        acc = __builtin_amdgcn_wmma_f32_16x16x4_f32(false, a1, false, w1,
                                                    (short)0, acc, false, false);
        v2f a2 = *(const v2f*)(Prow  + ko);
        v2f w2 = *(const v2f*)(W2row + ko);
        acc = __builtin_amdgcn_wmma_f32_16x16x4_f32(false, a2, false, w2,
                                                    (short)0, acc, false, false);
    }

    // epilogue: bias (with edge-norm scaling), LeakyReLU, partial row sumsq
    const int n = t * 16 + l15;        // output column
    float h[8];
#pragma unroll
    for (int r = 0; r < 8; ++r) {
        int   m  = r + 8 * half;       // row within 16-row group (C layout)
        float sv = Ss[g * 16 + m];
        float v  = acc[r] + b1s[n] * (1.0f + sv) + b2s[n] * sv;
        v = (v >= 0.0f) ? v : 0.2f * v;          // LeakyReLU(0.2)
        h[r] = v;
        float sq = v * v;
        sq += __shfl_xor(sq, 1, 16);             // reduce over the 16 cols this wave owns
        sq += __shfl_xor(sq, 2, 16);
        sq += __shfl_xor(sq, 4, 16);
        sq += __shfl_xor(sq, 8, 16);
        if (l15 == 0) atomicAdd(&rowss[g * 16 + m], sq);   // ds_add_f32 across 4 col-tiles
    }
    __syncthreads();

#pragma unroll
    for (int r = 0; r < 8; ++r) {
        int   m   = r + 8 * half;
        int   row = row0 + g * 16 + m;
        if (row >= Ntot) continue;
        float nrm = sqrtf(rowss[g * 16 + m]);
        float sc  = 1.0f / fmaxf(nrm, 1e-12f);
        __builtin_nontemporal_store(h[r] * sc, &out[(size_t)row * DD + n]);
    }
}

// ---------------------------------------------------------------------------
extern "C" void kernel_launch(void* const* d_in, const int* in_sizes, int n_in,
                              void* d_out, int out_size, void* d_ws, size_t ws_size,
                              hipStream_t stream)
{
    const float* feat_user = (const float*)d_in[0];
    const float* feat_item = (const float*)d_in[1];
    const float* W1        = (const float*)d_in[2];
    const float* b1        = (const float*)d_in[3];
    const float* W2        = (const float*)d_in[4];
    const float* b2        = (const float*)d_in[5];
    const float* norm_ui   = (const float*)d_in[6];
    const float* norm_iu   = (const float*)d_in[7];
    const int*   ui_src    = (const int*)  d_in[8];
    const int*   ui_dst    = (const int*)  d_in[9];
    float*       out       = (float*)d_out;

    const int NU   = in_sizes[0] / DD;
    const int NI   = in_sizes[1] / DD;
    const int E    = in_sizes[8];
    const int Ntot = NU + NI;

    // carve workspace: four D-wide aggregates + two scalar-sum arrays (~154 MB,
    // sized to stay L2-resident so the scatter atomics never touch HBM)
    float* aggU1 = (float*)d_ws;
    float* aggU2 = aggU1 + (size_t)NU * DD;
    float* aggI1 = aggU2 + (size_t)NU * DD;
    float* aggI2 = aggI1 + (size_t)NI * DD;
    float* sU    = aggI2 + (size_t)NI * DD;
    float* sI    = sU + NU;
    size_t used  = ((size_t)(NU + NI) * 2 * DD + NU + NI) * sizeof(float);
    hipMemsetAsync(d_ws, 0, used, stream);

    // phase 1: edge scatter (64 threads/edge)
    long threads = (long)E * DD;
    int  sblocks = (int)((threads + 255) / 256);
    ngcf_scatter<<<sblocks, 256, 0, stream>>>(
        feat_user, feat_item, norm_ui, norm_iu, ui_src, ui_dst,
        aggU1, aggU2, aggI1, aggI2, sU, sI, E);

    // phase 2: fused WMMA dual-GEMM + activation + row-normalize
    int gblocks = (Ntot + 31) / 32;
    ngcf_gemm<<<gblocks, 256, 0, stream>>>(
        feat_user, feat_item, W1, b1, W2, b2,
        aggU1, aggU2, aggI1, aggI2, sU, sI, out, NU, Ntot);
}